// DeepQNetIVCML_82729660055658
// MI455X (gfx1250) — compile-verified
//
#include <hip/hip_runtime.h>
#include <stdint.h>

// Problem constants (from reference)
#define N_NODES 512
#define DIM     768
#define NPAIR   128      // B*A = 4*32
#define QLEN    128      // L
#define ALPHA_F 0.8f

typedef __attribute__((ext_vector_type(8))) float v8f;
typedef __attribute__((ext_vector_type(2))) float v2f;

#if __has_builtin(__builtin_amdgcn_ballot_w32)
#define BALLOT32(p) __builtin_amdgcn_ballot_w32(p)
#else
#define BALLOT32(p) ((unsigned int)__ballot(p))
#endif

// ---------------------------------------------------------------------------
// Kernel 1: per-(b,a) graph propagation, bit-packed in LDS.
// Produces cw[pair][n] = coeff[n] / denom, where coeff[n] = alpha^(hop+1)
// at the first hop node n becomes covered (alpha^1 for initial set).
// Reads a_nei exactly once (coalesced 128B ballots).
// ---------------------------------------------------------------------------
__global__ __launch_bounds__(512) void prop_kernel(
    const float* __restrict__ a_nei, const float* __restrict__ vec_nei,
    float* __restrict__ cw) {
  __shared__ uint32_t adj[N_NODES * 16];   // 32 KB: adj[row][word]
  __shared__ uint32_t sbits[16];
  __shared__ uint32_t nxtbits[16];
  __shared__ float    coeff[N_NODES];
  __shared__ float    inv_denom;

  const int tid  = threadIdx.x;
  const int wave = tid >> 5, lane = tid & 31;
  const int pair = blockIdx.x;
  const float* Ap = a_nei + (size_t)pair * (N_NODES * (size_t)N_NODES);

  // Pack adjacency bits: wave w owns word-column w; 512 ballots per wave.
  for (int row = 0; row < N_NODES; ++row) {
    float v = Ap[(size_t)row * N_NODES + wave * 32 + lane];
    uint32_t m = (uint32_t)BALLOT32(v > 0.0f);
    if (lane == 0) adj[row * 16 + wave] = m;
  }
  if (wave == 0) {
    for (int w = 0; w < 16; ++w) {
      float v = vec_nei[pair * N_NODES + w * 32 + lane];
      uint32_t m = (uint32_t)BALLOT32(v > 0.0f);
      if (lane == 0) sbits[w] = m;
    }
    if (lane < 16) nxtbits[lane] = 0u;
  }
  __syncthreads();

  coeff[tid] = ((sbits[tid >> 5] >> (tid & 31)) & 1u) ? ALPHA_F : 0.0f;
  __syncthreads();

  const int w = tid & 15;
  const int g = tid >> 4;
  const float pw[3] = {0.64f, 0.512f, 0.4096f};  // alpha^2, alpha^3, alpha^4

  for (int step = 0; step < 3; ++step) {
    // nxt[o] = OR over rows i with s[i]=1 of adj[i][o]
    uint32_t acc = 0u;
    for (int r = g; r < N_NODES; r += 32) {
      if ((sbits[r >> 5] >> (r & 31)) & 1u) acc |= adj[r * 16 + w];
    }
    if (acc) atomicOr(&nxtbits[w], acc);
    __syncthreads();
    // newly covered nodes get this hop's coefficient (supports disjoint)
    {
      uint32_t sw = sbits[tid >> 5], nw = nxtbits[tid >> 5];
      uint32_t bit = 1u << (tid & 31);
      if (!(sw & bit) && (nw & bit)) coeff[tid] = pw[step];
    }
    __syncthreads();
    if (tid < 16) { sbits[tid] |= nxtbits[tid]; nxtbits[tid] = 0u; }
    __syncthreads();
  }

  if (tid == 0) {
    int d = 0;
    for (int i = 0; i < 16; ++i) d += __popc(sbits[i]);
    inv_denom = 1.0f / (float)d;   // denom==0 -> inf -> 0*inf=nan, matches 0/0
  }
  __syncthreads();
  cw[pair * N_NODES + tid] = coeff[tid] * inv_denom;
}

// ---------------------------------------------------------------------------
// Kernel 2: agg partials.  partial[pair][ns][d] = sum_{n in split} c[n]*F[n][d]
// fea_emb (192MB) read exactly once, fully coalesced over d.
// ---------------------------------------------------------------------------
__global__ __launch_bounds__(256) void wsum_kernel(
    const float* __restrict__ fea_emb, const float* __restrict__ cw,
    float* __restrict__ partial) {
  const int pair = blockIdx.z;
  const int ns   = blockIdx.y;                 // n-split: 4 x 128 rows
  const int d    = blockIdx.x * 256 + threadIdx.x;
  const float* F  = fea_emb + (size_t)pair * N_NODES * DIM + (size_t)ns * 128 * DIM + d;
  const float* cp = cw + pair * N_NODES + ns * 128;
  float acc = 0.0f;
#pragma unroll 4
  for (int n = 0; n < 128; ++n) acc += cp[n] * F[(size_t)n * DIM];
  partial[((pair * 4 + ns) * DIM) + d] = acc;
}

// Reduce 4 partials, apply nei_mask and relu (deterministic, no atomics).
__global__ __launch_bounds__(256) void aggfinish_kernel(
    const float* __restrict__ partial, const float* __restrict__ nei_mask,
    float* __restrict__ agg) {
  const int pair = blockIdx.y;
  const int d    = blockIdx.x * 256 + threadIdx.x;
  const float* p = partial + (size_t)pair * 4 * DIM + d;
  float v = p[0] + p[DIM] + p[2 * DIM] + p[3 * DIM];
  agg[pair * DIM + d] = (nei_mask[pair] > 0.0f) ? fmaxf(v, 0.0f) : 0.0f;
}

// qmean[b][d] = mean over L of query_fea[b][l][d]
__global__ __launch_bounds__(256) void qmean_kernel(
    const float* __restrict__ q, float* __restrict__ qm) {
  const int b = blockIdx.y;
  const int d = blockIdx.x * 256 + threadIdx.x;
  const float* p = q + (size_t)b * QLEN * DIM + d;
  float acc = 0.0f;
#pragma unroll 4
  for (int l = 0; l < QLEN; ++l) acc += p[(size_t)l * DIM];
  qm[b * DIM + d] = acc * (1.0f / (float)QLEN);
}

// Fill right half of H with broadcast q-mean: H[row][768+d] = qm[row/32][d]
__global__ __launch_bounds__(256) void hfill_kernel(
    const float* __restrict__ qm, float* __restrict__ H) {
  const int row = blockIdx.y;
  const int d   = blockIdx.x * 256 + threadIdx.x;
  H[(size_t)row * (2 * DIM) + DIM + d] = qm[(row >> 5) * DIM + d];
}

// ---------------------------------------------------------------------------
// fp32 WMMA GEMM: Out[M,N](ldo) = act(A[M,K] @ W[K,N] + bias), one wave per
// 16x16 tile, V_WMMA_F32_16X16X4_F32 K-loop.
// A frag (ISA 7.12.2, 32-bit A 16x4): lanes 0-15 rows M, v0/v1 = K0/K1;
// lanes 16-31 same rows, v0/v1 = K2/K3.  B mirrored with N across lanes.
// C/D: VGPR r holds M = r + 8*(lane>=16), N = lane%16.
// ---------------------------------------------------------------------------
__global__ __launch_bounds__(256) void gemm_wmma_kernel(
    const float* __restrict__ A, const float* __restrict__ W,
    const float* __restrict__ bias, float* __restrict__ Out,
    int M, int N, int K, int ldo, int relu) {
  const int wave = threadIdx.x >> 5;
  const int lane = threadIdx.x & 31;
  const int tile = blockIdx.x * 8 + wave;
  const int tiles_m = M >> 4;
  const int mtile = tile % tiles_m;
  const int ntile = tile / tiles_m;
  if (ntile >= (N >> 4)) return;   // wave-uniform

  const int half = lane >> 4;
  const int l16  = lane & 15;
  const float* arow = A + (size_t)(mtile * 16 + l16) * K;
  const float* wcol = W + (ntile * 16 + l16);

  v8f acc = {};
  for (int k0 = 0; k0 < K; k0 += 4) {
    const int ka = k0 + half * 2;
    v2f afrag;
    afrag.x = arow[ka];
    afrag.y = arow[ka + 1];
    v2f bfrag;
    bfrag.x = wcol[(size_t)ka * N];
    bfrag.y = wcol[(size_t)(ka + 1) * N];
    acc = __builtin_amdgcn_wmma_f32_16x16x4_f32(
        /*neg_a=*/false, afrag, /*neg_b=*/false, bfrag,
        /*c_mod=*/(short)0, acc, /*reuse_a=*/false, /*reuse_b=*/false);
  }

  const float bi = bias[ntile * 16 + l16];
#pragma unroll
  for (int r = 0; r < 8; ++r) {
    float v = acc[r] + bi;
    if (relu) v = fmaxf(v, 0.0f);
    Out[(size_t)(mtile * 16 + r + 8 * half) * ldo + (ntile * 16 + l16)] = v;
  }
}

// out[row] = dot(X2[row,:], W3) + b3   (one wave per row)
__global__ __launch_bounds__(32) void out_kernel(
    const float* __restrict__ X2, const float* __restrict__ W3,
    const float* __restrict__ b3, float* __restrict__ out) {
  const int row = blockIdx.x;
  const int lane = threadIdx.x;
  float acc = 0.0f;
#pragma unroll 4
  for (int j = lane; j < DIM; j += 32) acc += X2[(size_t)row * DIM + j] * W3[j];
  for (int off = 16; off > 0; off >>= 1) acc += __shfl_down(acc, off, 32);
  if (lane == 0) out[row] = acc + b3[0];
}

// ---------------------------------------------------------------------------
extern "C" void kernel_launch(void* const* d_in, const int* in_sizes, int n_in,
                              void* d_out, int out_size, void* d_ws, size_t ws_size,
                              hipStream_t stream) {
  const float* query_fea = (const float*)d_in[0];
  const float* a_nei     = (const float*)d_in[1];
  const float* vec_nei   = (const float*)d_in[2];
  const float* fea_emb   = (const float*)d_in[3];
  const float* nei_mask  = (const float*)d_in[4];
  const float* W1 = (const float*)d_in[5];
  const float* b1 = (const float*)d_in[6];
  const float* W2 = (const float*)d_in[7];
  const float* b2 = (const float*)d_in[8];
  const float* W3 = (const float*)d_in[9];
  const float* b3 = (const float*)d_in[10];
  float* out = (float*)d_out;

  // Workspace layout (floats): ~3.4 MB total
  float* ws   = (float*)d_ws;
  float* cw   = ws;                       // 128*512            = 65536
  float* qm   = cw + 65536;               // 4*768              = 3072
  float* part = qm + 3072;                // 128*4*768          = 393216
  float* agg  = part + 393216;            // 128*768            = 98304
  float* H    = agg + 98304;              // 128*1536           = 196608
  float* X2   = H + 196608;               // 128*768            = 98304

  // 1) propagation -> normalized per-node coefficients (reads a_nei once)
  prop_kernel<<<NPAIR, 512, 0, stream>>>(a_nei, vec_nei, cw);
  // 2) weighted sum over fea_emb (reads fea_emb once), deterministic reduce
  wsum_kernel<<<dim3(3, 4, NPAIR), 256, 0, stream>>>(fea_emb, cw, part);
  aggfinish_kernel<<<dim3(3, NPAIR), 256, 0, stream>>>(part, nei_mask, agg);
  // 3) query mean + MLP on fp32 WMMA
  qmean_kernel<<<dim3(3, 4), 256, 0, stream>>>(query_fea, qm);
  // X1 = relu(agg@W1+b1) written into H[:, :768]  (ldo = 1536)
  gemm_wmma_kernel<<<48, 256, 0, stream>>>(agg, W1, b1, H, NPAIR, DIM, DIM, 2 * DIM, 1);
  // H[:, 768:] = broadcast q-mean
  hfill_kernel<<<dim3(3, NPAIR), 256, 0, stream>>>(qm, H);
  // X2 = relu(H@W2+b2)   (K = 1536)
  gemm_wmma_kernel<<<48, 256, 0, stream>>>(H, W2, b2, X2, NPAIR, DIM, 2 * DIM, DIM, 1);
  // out = X2@W3 + b3
  out_kernel<<<NPAIR, 32, 0, stream>>>(X2, W3, b3, out);
}